// HorBlock_80058190397465
// MI455X (gfx1250) — compile-verified
//
#include <hip/hip_runtime.h>
#include <hip/hip_bf16.h>

typedef float v2f __attribute__((ext_vector_type(2)));
typedef float v8f __attribute__((ext_vector_type(8)));

#define WMMA_F32(a, b, c) \
    __builtin_amdgcn_wmma_f32_16x16x4_f32(false, (a), false, (b), (short)0, (c), false, false)

#define HW 1024
#define CDIM 384
#define NB 32
#define SUMD 744
#define FUSEDC 768

// ---------------------------------------------------------------------------
// K1: per-sample mean / rsqrt(var) over (C,H,W)
// ---------------------------------------------------------------------------
__global__ __launch_bounds__(256) void batch_stats(const float* __restrict__ x,
                                                   float* __restrict__ stats) {
  __shared__ float ssum[256];
  __shared__ float ssq[256];
  int b = blockIdx.x;
  const float* xb = x + (size_t)b * CDIM * HW;
  float s = 0.f, q = 0.f;
  for (int i = threadIdx.x; i < CDIM * HW; i += 256) {
    float v = xb[i];
    s += v;
    q += v * v;
  }
  ssum[threadIdx.x] = s;
  ssq[threadIdx.x] = q;
  __syncthreads();
  for (int o = 128; o > 0; o >>= 1) {
    if (threadIdx.x < o) {
      ssum[threadIdx.x] += ssum[threadIdx.x + o];
      ssq[threadIdx.x] += ssq[threadIdx.x + o];
    }
    __syncthreads();
  }
  if (threadIdx.x == 0) {
    float n = (float)(CDIM * HW);
    float mu = ssum[0] / n;
    float var = ssq[0] / n - mu * mu;
    stats[b * 2 + 0] = mu;
    stats[b * 2 + 1] = rsqrtf(var + 1e-5f);
  }
}

// ---------------------------------------------------------------------------
// K2: h = (x - mu) * rs * ln1_w[c,h,w] + ln1_b[c,h,w]
// ---------------------------------------------------------------------------
__global__ __launch_bounds__(256) void norm1(const float* __restrict__ x,
                                             const float* __restrict__ stats,
                                             const float* __restrict__ w,
                                             const float* __restrict__ bi,
                                             float* __restrict__ out) {
  size_t i = (size_t)blockIdx.x * 256 + threadIdx.x;
  size_t total = (size_t)NB * CDIM * HW;
  if (i >= total) return;
  int b = (int)(i / (CDIM * HW));
  int r = (int)(i % (CDIM * HW));
  float mu = stats[b * 2 + 0];
  float rs = stats[b * 2 + 1];
  out[i] = (x[i] - mu) * rs * w[r] + bi[r];
}

// ---------------------------------------------------------------------------
// Generic per-batch WMMA GEMM: out[b,m,p] = epi( sum_k W[m,k]*act[b,k,p] + bias[m] )
// grid = (ceil(M/64), 16, B), block = 256 (8 waves; wave -> 16x32 of the 64x64 tile)
// MODE 0: + bias           MODE 1: (+bias) * mul[b, mulOff+m, p]
// MODE 2: resid[b,m,p] + gamma[m] * (+bias)
// ---------------------------------------------------------------------------
template <int MODE>
__global__ __launch_bounds__(256) void gemm_bchw(
    const float* __restrict__ W, const float* __restrict__ act,
    const float* __restrict__ bias, float* __restrict__ out, int M, int K,
    const float* __restrict__ mul, int mulC, int mulOff,
    const float* __restrict__ resid, const float* __restrict__ gamma) {
  __shared__ float As[64][17];  // [m][k], pad 17 (17 coprime 64 -> no conflicts)
  __shared__ float Bs[16][65];  // [k][p]
  int b = blockIdx.z;
  int m0 = blockIdx.x * 64;
  int p0 = blockIdx.y * 64;
  int t = threadIdx.x;
  int lane = t & 31;
  int wave = t >> 5;
  int wm = wave & 3, wn = wave >> 2;
  int l16 = lane & 15, lhalf = lane >> 4;

  v8f acc0 = {0, 0, 0, 0, 0, 0, 0, 0};
  v8f acc1 = {0, 0, 0, 0, 0, 0, 0, 0};
  const float* actB = act + (size_t)b * K * HW;

  int kblocks = (K + 15) >> 4;
  for (int kb0 = 0; kb0 < kblocks; ++kb0) {
    int kb = kb0 * 16;
    {  // A tile 64x16: thread -> (m = t/4, 4 consecutive k)
      int m = t >> 2;
      int kq = t & 3;
      int k = kb + kq * 4;
      float4 v = make_float4(0.f, 0.f, 0.f, 0.f);
      if ((m0 + m) < M && (k + 4) <= K)
        v = *reinterpret_cast<const float4*>(W + (size_t)(m0 + m) * K + k);
      As[m][kq * 4 + 0] = v.x;
      As[m][kq * 4 + 1] = v.y;
      As[m][kq * 4 + 2] = v.z;
      As[m][kq * 4 + 3] = v.w;
    }
    {  // B tile 16x64: thread -> (k = t/16, 4 consecutive p)
      int k = t >> 4;
      int pq = t & 15;
      float4 v = make_float4(0.f, 0.f, 0.f, 0.f);
      if ((kb + k) < K)
        v = *reinterpret_cast<const float4*>(actB + (size_t)(kb + k) * HW + p0 + pq * 4);
      Bs[k][pq * 4 + 0] = v.x;
      Bs[k][pq * 4 + 1] = v.y;
      Bs[k][pq * 4 + 2] = v.z;
      Bs[k][pq * 4 + 3] = v.w;
    }
    __syncthreads();
#pragma unroll
    for (int s = 0; s < 4; ++s) {
      int k0 = s * 4 + lhalf * 2;  // A(M=l16, K=k0+v), v in {0,1}
      v2f a;
      a.x = As[wm * 16 + l16][k0];
      a.y = As[wm * 16 + l16][k0 + 1];
      v2f b0, b1;
      b0.x = Bs[k0][wn * 32 + l16];
      b0.y = Bs[k0 + 1][wn * 32 + l16];
      b1.x = Bs[k0][wn * 32 + 16 + l16];
      b1.y = Bs[k0 + 1][wn * 32 + 16 + l16];
      acc0 = WMMA_F32(a, b0, acc0);
      acc1 = WMMA_F32(a, b1, acc1);
    }
    __syncthreads();
  }

  size_t outB = (size_t)b * M * HW;
#pragma unroll
  for (int v = 0; v < 8; ++v) {
    int m = m0 + wm * 16 + v + 8 * lhalf;  // D(M = v + 8*(lane/16), N = lane%16)
    if (m >= M) continue;
    int pA = p0 + wn * 32 + l16;
    int pB = pA + 16;
    float r0 = acc0[v] + bias[m];
    float r1 = acc1[v] + bias[m];
    size_t o0 = outB + (size_t)m * HW + pA;
    size_t o1 = outB + (size_t)m * HW + pB;
    if (MODE == 0) {
      out[o0] = r0;
      out[o1] = r1;
    } else if (MODE == 1) {
      const float* mb = mul + ((size_t)b * mulC + mulOff + m) * HW;
      out[o0] = r0 * mb[pA];
      out[o1] = r1 * mb[pB];
    } else {
      out[o0] = resid[o0] + gamma[m] * r0;
      out[o1] = resid[o1] + gamma[m] * r1;
    }
  }
}

// ---------------------------------------------------------------------------
// Depthwise 7x7 (pad 3) on fused[:, 24:768], *1/3. One block per (b, ch) plane.
// ---------------------------------------------------------------------------
__global__ __launch_bounds__(256) void dwconv7(const float* __restrict__ fused,
                                               const float* __restrict__ w,
                                               const float* __restrict__ bias,
                                               float* __restrict__ out) {
  __shared__ float tile[38 * 38];
  __shared__ float wl[49];
  int bc = blockIdx.x;
  int ch = bc % SUMD;
  int b = bc / SUMD;
  const float* src = fused + ((size_t)b * FUSEDC + 24 + ch) * HW;
  int t = threadIdx.x;
  if (t < 49) wl[t] = w[ch * 49 + t];
  for (int i = t; i < 38 * 38; i += 256) {
    int r = i / 38, c = i % 38;
    int gy = r - 3, gx = c - 3;
    tile[i] = (gy >= 0 && gy < 32 && gx >= 0 && gx < 32) ? src[gy * 32 + gx] : 0.f;
  }
  __syncthreads();
  float bv = bias[ch];
  for (int p = t; p < HW; p += 256) {
    int y = p >> 5, xx = p & 31;
    float s = 0.f;
#pragma unroll
    for (int ky = 0; ky < 7; ++ky)
#pragma unroll
      for (int kx = 0; kx < 7; ++kx)
        s += wl[ky * 7 + kx] * tile[(y + ky) * 38 + (xx + kx)];
    out[((size_t)b * SUMD + ch) * HW + p] = (s + bv) * (1.0f / 3.0f);
  }
}

// ---------------------------------------------------------------------------
// y0 = fused[:, 0:24] * dw[:, 0:24]
// ---------------------------------------------------------------------------
__global__ __launch_bounds__(256) void y0_mul(const float* __restrict__ fused,
                                              const float* __restrict__ dw,
                                              float* __restrict__ y0) {
  int i = blockIdx.x * 256 + threadIdx.x;  // over 32*24*1024
  int b = i / (24 * HW);
  int r = i % (24 * HW);
  y0[i] = fused[(size_t)b * FUSEDC * HW + r] * dw[(size_t)b * SUMD * HW + r];
}

// ---------------------------------------------------------------------------
// Fused: per-pixel LN2 -> fc1 -> exact GELU -> fc2 -> x1 + gamma2*(.)
// grid = (HW/32, B); block 256. LDS: lnX 384x33, h1 chunk 64x33.
// Hidden (B*1536*HW = 201 MB) never touches HBM.
// ---------------------------------------------------------------------------
__global__ __launch_bounds__(256) void mlp_fused(
    const float* __restrict__ x1, const float* __restrict__ ln2_w,
    const float* __restrict__ ln2_b, const float* __restrict__ fc1_w,
    const float* __restrict__ fc1_b, const float* __restrict__ fc2_w,
    const float* __restrict__ fc2_b, const float* __restrict__ gamma2,
    float* __restrict__ out) {
  __shared__ float lnX[CDIM][33];
  __shared__ float h1buf[64][33];
  int b = blockIdx.y;
  int p0 = blockIdx.x * 32;
  int t = threadIdx.x;
  int lane = t & 31, wave = t >> 5;
  int l16 = lane & 15, lhalf = lane >> 4;
  const float* xb = x1 + (size_t)b * CDIM * HW;

  // Phase A: LN over channels; wave handles 4 pixels
  for (int j = 0; j < 4; ++j) {
    int pl = wave * 4 + j;
    int p = p0 + pl;
    float vals[12];
    float s = 0.f;
#pragma unroll
    for (int i = 0; i < 12; ++i) {
      vals[i] = xb[(size_t)(lane + 32 * i) * HW + p];
      s += vals[i];
    }
#pragma unroll
    for (int o = 16; o > 0; o >>= 1) s += __shfl_xor(s, o, 32);
    float mu = s * (1.0f / (float)CDIM);
    float q = 0.f;
#pragma unroll
    for (int i = 0; i < 12; ++i) {
      float d = vals[i] - mu;
      q += d * d;
    }
#pragma unroll
    for (int o = 16; o > 0; o >>= 1) q += __shfl_xor(q, o, 32);
    float rs = rsqrtf(q * (1.0f / (float)CDIM) + 1e-5f);
#pragma unroll
    for (int i = 0; i < 12; ++i) {
      int c = lane + 32 * i;
      lnX[c][pl] = (vals[i] - mu) * rs * ln2_w[c] + ln2_b[c];
    }
  }
  __syncthreads();

  v8f z = {0, 0, 0, 0, 0, 0, 0, 0};
  v8f accO[6];
#pragma unroll
  for (int i = 0; i < 6; ++i) accO[i] = z;

  int tm = wave & 3, tn = wave >> 2;
  for (int chunk = 0; chunk < 24; ++chunk) {
    int o0 = chunk * 64;
    // Phase B: H1 tile = gelu(fc1_w[chunk] @ lnX + b); one 16x16 per wave
    v8f acc = z;
    const float* wrow = fc1_w + (size_t)(o0 + tm * 16 + l16) * CDIM;
    for (int kb = 0; kb < CDIM; kb += 4) {
      int kk = kb + 2 * lhalf;
      v2f a;
      a.x = wrow[kk];
      a.y = wrow[kk + 1];
      v2f bb;
      bb.x = lnX[kk][tn * 16 + l16];
      bb.y = lnX[kk + 1][tn * 16 + l16];
      acc = WMMA_F32(a, bb, acc);
    }
    __syncthreads();  // prior Phase C reads of h1buf finished
#pragma unroll
    for (int v = 0; v < 8; ++v) {
      int ol = tm * 16 + v + 8 * lhalf;
      float hv = acc[v] + fc1_b[o0 + ol];
      h1buf[ol][tn * 16 + l16] =
          0.5f * hv * (1.0f + erff(hv * 0.70710678118654752f));
    }
    __syncthreads();
    // Phase C: accO += fc2_w[:, chunk] @ h1 ; 6 of 48 tiles per wave
#pragma unroll
    for (int i = 0; i < 6; ++i) {
      int tid = wave * 6 + i;
      int dt = tid >> 1, pt = tid & 1;
      const float* w2row = fc2_w + (size_t)(dt * 16 + l16) * 1536 + o0;
#pragma unroll
      for (int kb = 0; kb < 64; kb += 4) {
        int kk = kb + 2 * lhalf;
        v2f a;
        a.x = w2row[kk];
        a.y = w2row[kk + 1];
        v2f bb;
        bb.x = h1buf[kk][pt * 16 + l16];
        bb.y = h1buf[kk + 1][pt * 16 + l16];
        accO[i] = WMMA_F32(a, bb, accO[i]);
      }
    }
  }
  __syncthreads();
#pragma unroll
  for (int i = 0; i < 6; ++i) {
    int tid = wave * 6 + i;
    int dt = tid >> 1, pt = tid & 1;
#pragma unroll
    for (int v = 0; v < 8; ++v) {
      int d = dt * 16 + v + 8 * lhalf;
      int p = p0 + pt * 16 + l16;
      float r = accO[i][v] + fc2_b[d];
      out[(size_t)b * CDIM * HW + (size_t)d * HW + p] =
          xb[(size_t)d * HW + p] + gamma2[d] * r;
    }
  }
}

// ---------------------------------------------------------------------------
extern "C" void kernel_launch(void* const* d_in, const int* in_sizes, int n_in,
                              void* d_out, int out_size, void* d_ws,
                              size_t ws_size, hipStream_t stream) {
  (void)in_sizes; (void)n_in; (void)out_size; (void)ws_size;
  const float* x = (const float*)d_in[0];
  const float* ln1_w = (const float*)d_in[1];
  const float* ln1_b = (const float*)d_in[2];
  const float* proj_in_w = (const float*)d_in[3];
  const float* proj_in_b = (const float*)d_in[4];
  const float* dw_w = (const float*)d_in[5];
  const float* dw_b = (const float*)d_in[6];
  const float* pw_w[4] = {(const float*)d_in[7], (const float*)d_in[9],
                          (const float*)d_in[11], (const float*)d_in[13]};
  const float* pw_b[4] = {(const float*)d_in[8], (const float*)d_in[10],
                          (const float*)d_in[12], (const float*)d_in[14]};
  const float* proj_out_w = (const float*)d_in[15];
  const float* proj_out_b = (const float*)d_in[16];
  const float* ln2_w = (const float*)d_in[17];
  const float* ln2_b = (const float*)d_in[18];
  const float* fc1_w = (const float*)d_in[19];
  const float* fc1_b = (const float*)d_in[20];
  const float* fc2_w = (const float*)d_in[21];
  const float* fc2_b = (const float*)d_in[22];
  const float* gamma1 = (const float*)d_in[23];
  const float* gamma2 = (const float*)d_in[24];
  float* outp = (float*)d_out;

  // workspace layout (bytes): [stats 1KB][bufH SH][bufF 2*SH][bufD 744/384*SH]
  char* ws = (char*)d_ws;
  const size_t SH = (size_t)NB * CDIM * HW * 4;  // 50.33 MB
  float* stats = (float*)ws;
  float* bufH = (float*)(ws + 1024);
  float* bufF = (float*)(ws + 1024 + SH);
  float* bufD = (float*)(ws + 1024 + 3 * SH);
  float* yA = bufH;
  float* yB = (float*)((char*)bufH + SH / 2);
  float* y4 = bufF;            // fused dead by then
  float* x1 = (float*)((char*)bufF + SH);

  batch_stats<<<NB, 256, 0, stream>>>(x, stats);
  norm1<<<(NB * CDIM * HW) / 256, 256, 0, stream>>>(x, stats, ln1_w, ln1_b, bufH);
  // proj_in: 768 x 384
  gemm_bchw<0><<<dim3(12, 16, NB), 256, 0, stream>>>(
      proj_in_w, bufH, proj_in_b, bufF, 768, 384, nullptr, 0, 0, nullptr, nullptr);
  dwconv7<<<NB * SUMD, 256, 0, stream>>>(bufF, dw_w, dw_b, bufD);
  y0_mul<<<(NB * 24 * HW) / 256, 256, 0, stream>>>(bufF, bufD, yA);
  // gated pointwise chain (mul offsets = cumsum of DIMS)
  gemm_bchw<1><<<dim3(1, 16, NB), 256, 0, stream>>>(
      pw_w[0], yA, pw_b[0], yB, 48, 24, bufD, SUMD, 24, nullptr, nullptr);
  gemm_bchw<1><<<dim3(2, 16, NB), 256, 0, stream>>>(
      pw_w[1], yB, pw_b[1], yA, 96, 48, bufD, SUMD, 72, nullptr, nullptr);
  gemm_bchw<1><<<dim3(3, 16, NB), 256, 0, stream>>>(
      pw_w[2], yA, pw_b[2], yB, 192, 96, bufD, SUMD, 168, nullptr, nullptr);
  gemm_bchw<1><<<dim3(6, 16, NB), 256, 0, stream>>>(
      pw_w[3], yB, pw_b[3], y4, 384, 192, bufD, SUMD, 360, nullptr, nullptr);
  // proj_out + gamma1 residual -> x1
  gemm_bchw<2><<<dim3(6, 16, NB), 256, 0, stream>>>(
      proj_out_w, y4, proj_out_b, x1, 384, 384, nullptr, 0, 0, x, gamma1);
  // LN2 + MLP fused -> out
  mlp_fused<<<dim3(HW / 32, NB), 256, 0, stream>>>(
      x1, ln2_w, ln2_b, fc1_w, fc1_b, fc2_w, fc2_b, gamma2, outp);
}